// CoSaMP_layer_23270132810501
// MI455X (gfx1250) — compile-verified
//
#include <hip/hip_runtime.h>
#include <hip/hip_bf16.h>

typedef __attribute__((ext_vector_type(2))) float v2f;
typedef __attribute__((ext_vector_type(4))) float v4f;
typedef __attribute__((ext_vector_type(8))) float v8f;

#define N_DIM 8192
#define M_DIM 2048
#define B_DIM 256
#define S2 128          // 2*s candidate support
#define SP 64           // s
#define PITCH 129       // LDS pitch for 128x128 Gram
#define KCH 32          // K chunk staged in LDS for the GEMMs
#define PK 36           // LDS pitch for [row][k] chunks (conflict-free, 16B align)

__device__ __forceinline__ v8f wmma4(v2f a, v2f b, v8f c) {
  // D = A(16x4,f32) * B(4x16,f32) + C(16x16,f32)
  return __builtin_amdgcn_wmma_f32_16x16x4_f32(false, a, false, b, (short)0, c,
                                               false, false);
}

// ---------------------------------------------------------------- zero output
__global__ __launch_bounds__(256) void zero_kernel(float* p, int n) {
  int i = blockIdx.x * blockDim.x + threadIdx.x;
  int stride = gridDim.x * blockDim.x;
  for (; i < n; i += stride) p[i] = 0.0f;
}

// --------------------------------------------------- R = Y - (A @ X)^T  [B,M]
// D[b][m] = sum_n X[n][b] * A[m][n];  R = Y - D
// Block tile: 64(b) x 64(m), 4 waves (one 16-b strip each) sharing the A chunk
// staged in LDS.  Grid: 4 b-groups x 32 m-groups = 128 blocks.
__global__ __launch_bounds__(128) void gemm_r_kernel(const float* __restrict__ A,
                                                     const float* __restrict__ X,
                                                     const float* __restrict__ Y,
                                                     float* __restrict__ R) {
  __shared__ float Ash[64 * PK];                 // [m-row][k], 9.2 KB

  const int tid = threadIdx.x;
  const int lane = tid & 31;
  const int w = tid >> 5;                        // 0..3
  const int bh = blockIdx.x & 3;                 // 4 groups of 64 b-rows
  const int mg = blockIdx.x >> 2;                // 32 m-groups of width 64
  const int b0 = bh * 64 + 16 * w;
  const int m0 = mg * 64;
  const int l16 = lane & 15;
  const int kh = (lane >> 4) << 1;               // 2*(lane/16)
  const int rowhalf = (lane >> 4) << 3;          // 8*(lane/16)

  v8f acc[4];
#pragma unroll
  for (int t = 0; t < 4; ++t)
#pragma unroll
    for (int r = 0; r < 8; ++r) acc[t][r] = 0.0f;

  for (int k0 = 0; k0 < N_DIM; k0 += KCH) {
    // stage A[m0:m0+64, k0:k0+32] -> Ash[row*PK + kk] (coalesced float4 x2)
#pragma unroll
    for (int e = tid; e < 256; e += 128) {
      const int row = e >> 2;
      const int ks = (e & 3) * 8;
      const float* gp = A + (size_t)(m0 + row) * N_DIM + k0 + ks;
      *reinterpret_cast<v4f*>(&Ash[row * PK + ks])     = *reinterpret_cast<const v4f*>(gp);
      *reinterpret_cast<v4f*>(&Ash[row * PK + ks + 4]) = *reinterpret_cast<const v4f*>(gp + 4);
    }
    __syncthreads();
#pragma unroll
    for (int kk = 0; kk < KCH; kk += 4) {
      v2f af;
      af.x = X[(size_t)(k0 + kk + kh) * B_DIM + b0 + l16];
      af.y = X[(size_t)(k0 + kk + kh + 1) * B_DIM + b0 + l16];
#pragma unroll
      for (int t = 0; t < 4; ++t) {
        const v2f bf = *reinterpret_cast<const v2f*>(
            &Ash[(16 * t + l16) * PK + kk + kh]);
        acc[t] = wmma4(af, bf, acc[t]);
      }
    }
    __syncthreads();
  }
#pragma unroll
  for (int t = 0; t < 4; ++t)
#pragma unroll
    for (int r = 0; r < 8; ++r) {
      const int bb = b0 + r + rowhalf;
      const int mm = m0 + 16 * t + l16;
      R[(size_t)bb * M_DIM + mm] = Y[(size_t)bb * M_DIM + mm] - acc[t][r];
    }
}

// ------------------------------------------------ proxy = |R @ A|  [B,N]
// Block tile: 128(b) x 64(n), 8 waves sharing the A chunk (transposed to
// [n][k] in LDS so both fragment halves are one ds_load_b64).
// Grid: 2 b-halves x 128 n-groups = 256 blocks.
__global__ __launch_bounds__(256) void gemm_proxy_kernel(const float* __restrict__ R,
                                                         const float* __restrict__ A,
                                                         float* __restrict__ proxy) {
  __shared__ float Ash[64 * PK];                 // [n-row][k], 9.2 KB

  const int tid = threadIdx.x;
  const int lane = tid & 31;
  const int w = tid >> 5;                        // 0..7
  const int bh = blockIdx.x & 1;                 // 2 halves of 128 b-rows
  const int ng = blockIdx.x >> 1;                // 128 n-groups of width 64
  const int b0 = bh * 128 + 16 * w;
  const int n0 = ng * 64;
  const int l16 = lane & 15;
  const int kh = (lane >> 4) << 1;
  const int rowhalf = (lane >> 4) << 3;

  v8f acc[4];
#pragma unroll
  for (int t = 0; t < 4; ++t)
#pragma unroll
    for (int r = 0; r < 8; ++r) acc[t][r] = 0.0f;

  for (int k0 = 0; k0 < M_DIM; k0 += KCH) {
    // stage A[k0:k0+32, n0:n0+64] transposed -> Ash[nn*PK + kk]
#pragma unroll
    for (int e = tid; e < 2048; e += 256) {
      const int kk = e >> 6;                     // 0..31
      const int nn = e & 63;                     // coalesced along n
      Ash[nn * PK + kk] = A[(size_t)(k0 + kk) * N_DIM + n0 + nn];
    }
    __syncthreads();
#pragma unroll
    for (int kk = 0; kk < KCH; kk += 4) {
      const v2f af = *reinterpret_cast<const v2f*>(
          R + (size_t)(b0 + l16) * M_DIM + k0 + kk + kh);
#pragma unroll
      for (int t = 0; t < 4; ++t) {
        const v2f bf = *reinterpret_cast<const v2f*>(
            &Ash[(16 * t + l16) * PK + kk + kh]);
        acc[t] = wmma4(af, bf, acc[t]);
      }
    }
    __syncthreads();
  }
#pragma unroll
  for (int t = 0; t < 4; ++t)
#pragma unroll
    for (int r = 0; r < 8; ++r) {
      const int bb = b0 + r + rowhalf;
      const int nn = n0 + 16 * t + l16;
      proxy[(size_t)bb * N_DIM + nn] = fabsf(acc[t][r]);
    }
}

// -------------------------------------------------- top-128 per row of proxy
__global__ __launch_bounds__(256) void topk_kernel(const float* __restrict__ proxy,
                                                   int* __restrict__ idx) {
  __shared__ float vals[N_DIM];
  __shared__ float rv[256];
  __shared__ int ri[256];
  __shared__ float sv[256];
  __shared__ int si[256];
  __shared__ int winJ;

  const int b = blockIdx.x;
  const int tid = threadIdx.x;
  const float* row = proxy + (size_t)b * N_DIM;

  for (int j = tid; j < N_DIM; j += 256) vals[j] = row[j];
  __syncthreads();

  float lm = -1.0f; int li = 0;
  for (int j = tid; j < N_DIM; j += 256)
    if (vals[j] > lm) { lm = vals[j]; li = j; }
  rv[tid] = lm; ri[tid] = li;
  __syncthreads();

  for (int it = 0; it < S2; ++it) {
    sv[tid] = rv[tid]; si[tid] = ri[tid];
    __syncthreads();
#pragma unroll
    for (int s = 128; s > 0; s >>= 1) {
      if (tid < s && sv[tid + s] > sv[tid]) {
        sv[tid] = sv[tid + s]; si[tid] = si[tid + s];
      }
      __syncthreads();
    }
    if (tid == 0) { idx[b * S2 + it] = si[0]; winJ = si[0]; }
    __syncthreads();
    const int j = winJ;
    if ((j & 255) == tid) {            // owner refreshes its cached max
      vals[j] = -1.0f;
      float m2 = -1.0f; int i2 = 0;
      for (int jj = tid; jj < N_DIM; jj += 256)
        if (vals[jj] > m2) { m2 = vals[jj]; i2 = jj; }
      rv[tid] = m2; ri[tid] = i2;
    }
    __syncthreads();
  }
}

// ------------- per-sample: Gram (WMMA) + rhs, SPD solve, top-64, scatter ----
extern __shared__ float smem[];

__global__ __launch_bounds__(256) void solve_kernel(const float* __restrict__ A,
                                                    const float* __restrict__ Y,
                                                    const int* __restrict__ idx,
                                                    float* __restrict__ out) {
  float* G    = smem;                    // 128*129
  float* As   = G + S2 * PITCH;          // 32*129
  float* Ysh  = As + 32 * PITCH;         // 32
  float* bv   = Ysh + 32;                // 128
  float* u    = bv + S2;                 // 128
  float* absu = u + S2;                  // 128
  int*   idxsh = (int*)(absu + S2);      // 128
  int*   sel   = idxsh + S2;             // 64

  const int b = blockIdx.x;
  const int tid = threadIdx.x;
  const int lane = tid & 31;
  const int w = tid >> 5;
  const int l16 = lane & 15;
  const int kh = (lane >> 4) << 1;
  const int rowhalf = (lane >> 4) << 3;

  if (tid < S2) idxsh[tid] = idx[b * S2 + tid];
  __syncthreads();

  v8f accG[8];
#pragma unroll
  for (int t = 0; t < 8; ++t)
#pragma unroll
    for (int r = 0; r < 8; ++r) accG[t][r] = 0.0f;
  float accb = 0.0f;

  for (int k0 = 0; k0 < M_DIM; k0 += 32) {
    if (tid < 32) Ysh[tid] = Y[(size_t)b * M_DIM + k0 + tid];
    for (int e = tid; e < 32 * S2; e += 256) {
      const int kk = e >> 7, i = e & 127;
      As[kk * PITCH + i] = A[(size_t)(k0 + kk) * N_DIM + idxsh[i]];
    }
    __syncthreads();
    if (tid < S2) {
#pragma unroll 4
      for (int kk = 0; kk < 32; ++kk) accb += As[kk * PITCH + tid] * Ysh[kk];
    }
#pragma unroll
    for (int kt = 0; kt < 8; ++kt) {
      const int kk = kt * 4 + kh;
      v2f af;
      af.x = As[kk * PITCH + 16 * w + l16];
      af.y = As[(kk + 1) * PITCH + 16 * w + l16];
#pragma unroll
      for (int t = 0; t < 8; ++t) {
        v2f bf;
        bf.x = As[kk * PITCH + 16 * t + l16];
        bf.y = As[(kk + 1) * PITCH + 16 * t + l16];
        accG[t] = wmma4(af, bf, accG[t]);
      }
    }
    __syncthreads();
  }

  // spill G strip (this wave owns rows [16w, 16w+16)) and rhs into LDS
#pragma unroll
  for (int t = 0; t < 8; ++t)
#pragma unroll
    for (int r = 0; r < 8; ++r)
      G[(16 * w + r + rowhalf) * PITCH + 16 * t + l16] = accG[t][r];
  if (tid < S2) bv[tid] = accb;
  __syncthreads();

  // Gaussian elimination (G is SPD -> no pivoting needed)
  for (int col = 0; col < S2; ++col) {
    const float inv = 1.0f / G[col * PITCH + col];
    const int nr = S2 - 1 - col;
    for (int e = tid; e < nr * nr; e += 256) {
      const int r = col + 1 + e / nr;
      const int j = col + 1 + e % nr;
      G[r * PITCH + j] -= G[r * PITCH + col] * inv * G[col * PITCH + j];
    }
    for (int r = col + 1 + tid; r < S2; r += 256)
      bv[r] -= G[r * PITCH + col] * inv * bv[col];
    __syncthreads();
  }
  // back substitution (column-oriented)
  for (int col = S2 - 1; col >= 0; --col) {
    if (tid == 0) u[col] = bv[col] / G[col * PITCH + col];
    __syncthreads();
    const float uc = u[col];
    for (int r = tid; r < col; r += 256) bv[r] -= G[r * PITCH + col] * uc;
    __syncthreads();
  }

  // top-64 of |u| over the 128 candidates (serial in thread 0: 64*128 ops)
  if (tid < S2) absu[tid] = fabsf(u[tid]);
  __syncthreads();
  if (tid == 0) {
    for (int it = 0; it < SP; ++it) {
      float best = -1.0f; int bi = 0;
      for (int i = 0; i < S2; ++i)
        if (absu[i] > best) { best = absu[i]; bi = i; }
      sel[it] = bi;
      absu[bi] = -1.0f;
    }
  }
  __syncthreads();

  // scatter into H [N,B] and S_out [N,B] (pre-zeroed)
  for (int t = tid; t < SP; t += 256) {
    const int j = sel[t];
    const int gi = idxsh[j];
    out[(size_t)gi * B_DIM + b] = u[j];
    out[(size_t)N_DIM * B_DIM + (size_t)gi * B_DIM + b] = 1.0f;
  }
}

// ---------------------------------------------------------------------------
extern "C" void kernel_launch(void* const* d_in, const int* in_sizes, int n_in,
                              void* d_out, int out_size, void* d_ws, size_t ws_size,
                              hipStream_t stream) {
  const float* X = (const float*)d_in[0];   // [N, B]
  const float* Y = (const float*)d_in[1];   // [B, M]
  // d_in[2] = S (empty incoming support) is unused on the first iteration
  const float* A = (const float*)d_in[3];   // [M, N]
  float* out = (float*)d_out;               // H [N,B] then S_out [N,B]

  float* R     = (float*)d_ws;              // [B, M]   524288 f
  float* proxy = R + (size_t)B_DIM * M_DIM; // [B, N]  2097152 f
  int*   idx   = (int*)(proxy + (size_t)B_DIM * N_DIM); // [B, 128]

  zero_kernel<<<2048, 256, 0, stream>>>(out, out_size);
  gemm_r_kernel<<<128, 128, 0, stream>>>(A, X, Y, R);
  gemm_proxy_kernel<<<256, 256, 0, stream>>>(R, A, proxy);
  topk_kernel<<<256, 256, 0, stream>>>(proxy, idx);

  const size_t smem_bytes =
      (size_t)(S2 * PITCH + 32 * PITCH + 32 + S2 + S2 + S2 + S2 + SP) * 4;
  solve_kernel<<<256, 256, smem_bytes, stream>>>(A, Y, idx, out);
}